// FGN_layer_37933151158647
// MI455X (gfx1250) — compile-verified
//
#include <hip/hip_runtime.h>
#include <hip/hip_bf16.h>
#include <math.h>

typedef __attribute__((ext_vector_type(2))) float v2f;
typedef __attribute__((ext_vector_type(8))) float v8f;

#define IN_F   768
#define OUT_F  1024
#define BATCH  512
#define LSTR   20              // LDS row stride in floats: 80B -> 16B aligned, conflict-free columns
#define EPS    1e-3f
#define TILE_FLOATS (64 * LSTR)        // one 64x16 tile (padded)
#define BUF_FLOATS  (3 * TILE_FLOATS)  // x + W + C tiles
#define NCHUNK      (IN_F / 16)

// s_wait_asynccnt: builtin if the toolchain has it, else raw ISA
#if defined(__has_builtin) && __has_builtin(__builtin_amdgcn_s_wait_asynccnt)
#define WAIT_ASYNC(n) __builtin_amdgcn_s_wait_asynccnt(n)
#else
#define WAIT_ASYNC(n) asm volatile("s_wait_asynccnt %0" :: "i"(n) : "memory")
#endif

// Async global->LDS B128 copy (CDNA5, tracked by ASYNCcnt).
// lds_byte_off: wave-relative LDS byte offset (low 32 bits of the flat shared addr).
static __device__ __forceinline__ void async_b128(unsigned lds_byte_off, const float* gptr) {
    asm volatile("global_load_async_to_lds_b128 %0, %1, off"
                 :: "v"(lds_byte_off), "v"(gptr)
                 : "memory");
}

// ---------------------------------------------------------------------------
// Pre-pass: one wave per row.
//   wid <  OUT_F          : bias[o] = -sum(W[o]*C[o]),  cnorm[o] = sum(C[o]^2)
//   wid in [OUT_F, +BATCH): xnorm[b] = sum(x[b]^2)
// ---------------------------------------------------------------------------
__global__ __launch_bounds__(256) void fgn_pre(const float* __restrict__ W,
                                               const float* __restrict__ C,
                                               const float* __restrict__ x,
                                               float* __restrict__ bias,
                                               float* __restrict__ cnorm,
                                               float* __restrict__ xnorm) {
    const int lane = threadIdx.x & 31;
    const int wid  = (blockIdx.x * blockDim.x + threadIdx.x) >> 5;
    if (wid < OUT_F) {
        const float* wr = W + (size_t)wid * IN_F;
        const float* cr = C + (size_t)wid * IN_F;
        float swc = 0.0f, scc = 0.0f;
        for (int i = lane; i < IN_F; i += 32) {
            const float wv = wr[i], cv = cr[i];
            swc = fmaf(wv, cv, swc);
            scc = fmaf(cv, cv, scc);
        }
        for (int off = 16; off > 0; off >>= 1) {
            swc += __shfl_down(swc, off, 32);
            scc += __shfl_down(scc, off, 32);
        }
        if (lane == 0) { bias[wid] = -swc; cnorm[wid] = scc; }
    } else if (wid < OUT_F + BATCH) {
        const int r = wid - OUT_F;
        const float* xr = x + (size_t)r * IN_F;
        float s = 0.0f;
        for (int i = lane; i < IN_F; i += 32) { const float v = xr[i]; s = fmaf(v, v, s); }
        for (int off = 16; off > 0; off >>= 1) s += __shfl_down(s, off, 32);
        if (lane == 0) xnorm[r] = s;
    }
}

// ---------------------------------------------------------------------------
// softmax(pis) over OUT_F elements; also zero the scalar NLL output slot.
// ---------------------------------------------------------------------------
__global__ __launch_bounds__(256) void fgn_softmax(const float* __restrict__ pis,
                                                   float* __restrict__ pisn,
                                                   float* __restrict__ nll_out) {
    __shared__ float red[256];
    const int t = threadIdx.x;
    float v0 = pis[t], v1 = pis[t + 256], v2 = pis[t + 512], v3 = pis[t + 768];
    float mx = fmaxf(fmaxf(v0, v1), fmaxf(v2, v3));
    red[t] = mx;
    __syncthreads();
    for (int s = 128; s > 0; s >>= 1) {
        if (t < s) red[t] = fmaxf(red[t], red[t + s]);
        __syncthreads();
    }
    mx = red[0];
    __syncthreads();
    const float e0 = __expf(v0 - mx), e1 = __expf(v1 - mx);
    const float e2 = __expf(v2 - mx), e3 = __expf(v3 - mx);
    red[t] = e0 + e1 + e2 + e3;
    __syncthreads();
    for (int s = 128; s > 0; s >>= 1) {
        if (t < s) red[t] += red[t + s];
        __syncthreads();
    }
    const float inv = 1.0f / red[0];
    pisn[t]       = e0 * inv;
    pisn[t + 256] = e1 * inv;
    pisn[t + 512] = e2 * inv;
    pisn[t + 768] = e3 * inv;
    if (t == 0) nll_out[0] = 0.0f;
}

// ---------------------------------------------------------------------------
// Main fused kernel: 64(M)x64(N) block tile, 8 waves (wave32).
// Wave (wm = w&3, wn = w>>2) owns rows [wm*16,+16), cols [wn*32,+32):
// two 16x16 fp32 WMMA tiles, each with a W-accumulator and a C-accumulator.
// K streamed in chunks of 16 through DOUBLE-BUFFERED LDS filled with
// global_load_async_to_lds_b128 (ASYNCcnt), next chunk in flight during compute.
// ---------------------------------------------------------------------------
__global__ __launch_bounds__(256) void fgn_main(const float* __restrict__ x,
                                                const float* __restrict__ W,
                                                const float* __restrict__ C,
                                                const float* __restrict__ sigs,
                                                const float* __restrict__ bias,
                                                const float* __restrict__ cnorm,
                                                const float* __restrict__ xnorm,
                                                const float* __restrict__ pisn,
                                                float* __restrict__ res,
                                                float* __restrict__ nll_out) {
    __shared__ float smem[2 * BUF_FLOATS];

    const int tid  = threadIdx.x;
    const int lane = tid & 31;
    const int wave = tid >> 5;
    const int m0   = blockIdx.y * 64;
    const int n0   = blockIdx.x * 64;

    const int wm = wave & 3;   // M strip: wm*16
    const int wn = wave >> 2;  // N strip: wn*32

    v8f accW0 = {}; v8f accW1 = {};
    v8f accC0 = {}; v8f accC1 = {};

    // chunk-loader indices: thread t fills row t/4, 16B at column (t%4)*4
    const int lrow  = tid >> 2;
    const int lcol4 = (tid & 3) << 2;

    // LDS byte offsets for the async fills (low 32 bits of flat shared addr ==
    // wave-relative LDS byte offset per the ISA LDS-aperture mapping)
    const unsigned lds_base = (unsigned)(uintptr_t)smem;
    const unsigned st_off   = (unsigned)((lrow * LSTR + lcol4) * 4);

    // per-thread global source rows (constant across chunks)
    const float* gx = x + (size_t)(m0 + lrow) * IN_F + lcol4;
    const float* gw = W + (size_t)(n0 + lrow) * IN_F + lcol4;
    const float* gc = C + (size_t)(n0 + lrow) * IN_F + lcol4;

    // WMMA fragment offsets within one buffer (fp32 16x16x4 layout):
    //   lanes 0-15 carry K = k,k+1 ; lanes 16-31 carry K = k+2,k+3
    const int fr = lane & 15;
    const int fk = (lane >> 4) << 1;                          // 0 or 2
    const int aOff  = (wm * 16 + fr) * LSTR + fk;                          // in lx
    const int b0Off = TILE_FLOATS     + (wn * 32 + fr)      * LSTR + fk;   // in lw
    const int b1Off = TILE_FLOATS     + (wn * 32 + 16 + fr) * LSTR + fk;
    const int c0Off = 2 * TILE_FLOATS + (wn * 32 + fr)      * LSTR + fk;   // in lc
    const int c1Off = 2 * TILE_FLOATS + (wn * 32 + 16 + fr) * LSTR + fk;

    // prologue: fill buffer 0 with chunk 0 (3 async B128 per thread)
    {
        const unsigned b = lds_base + st_off;
        async_b128(b,                        gx);
        async_b128(b + TILE_FLOATS * 4,      gw);
        async_b128(b + 2 * TILE_FLOATS * 4,  gc);
    }

    for (int i = 0; i < NCHUNK; ++i) {
        const int cur = i & 1;
        if (i + 1 < NCHUNK) {
            // issue next chunk into the other buffer while this one computes
            const int kc = (i + 1) * 16;
            const unsigned b = lds_base + (unsigned)(1 - cur) * (BUF_FLOATS * 4) + st_off;
            async_b128(b,                        gx + kc);
            async_b128(b + TILE_FLOATS * 4,      gw + kc);
            async_b128(b + 2 * TILE_FLOATS * 4,  gc + kc);
            WAIT_ASYNC(3);   // in-order completion -> chunk i has landed
        } else {
            WAIT_ASYNC(0);
        }
        __syncthreads();     // chunk i visible to every wave

        const float* buf = smem + (cur ? BUF_FLOATS : 0);
#pragma unroll
        for (int ks = 0; ks < 16; ks += 4) {
            const v2f a   = *(const v2f*)(buf + aOff  + ks);
            const v2f bw0 = *(const v2f*)(buf + b0Off + ks);
            const v2f bw1 = *(const v2f*)(buf + b1Off + ks);
            const v2f bc0 = *(const v2f*)(buf + c0Off + ks);
            const v2f bc1 = *(const v2f*)(buf + c1Off + ks);
            accW0 = __builtin_amdgcn_wmma_f32_16x16x4_f32(false, a, false, bw0, (short)0, accW0, false, false);
            accW1 = __builtin_amdgcn_wmma_f32_16x16x4_f32(false, a, false, bw1, (short)0, accW1, false, false);
            accC0 = __builtin_amdgcn_wmma_f32_16x16x4_f32(false, a, false, bc0, (short)0, accC0, false, false);
            accC1 = __builtin_amdgcn_wmma_f32_16x16x4_f32(false, a, false, bc1, (short)0, accC1, false, false);
        }
        __syncthreads();     // everyone done reading buf[cur] before it is refilled
    }

    // Epilogue. C/D layout: lane 0-15 -> N=lane, M=vgpr; lane 16-31 -> N=lane-16, M=vgpr+8.
    const float inv_total = 1.0f / ((float)BATCH * (float)OUT_F);
    const int row_base = m0 + wm * 16 + ((lane >> 4) << 3);
    float partial = 0.0f;

#pragma unroll
    for (int tn = 0; tn < 2; ++tn) {
        const int n = n0 + wn * 32 + tn * 16 + (lane & 15);
        const float bs = bias[n];
        const float cn = cnorm[n];
        const float sg = sigs[n];
        const float sgc = fmaxf(sg, EPS);
        const float inv_s2c = 1.0f / (sgc * sgc);
        const float inv_s2  = 1.0f / (sg * sg);
        const float logterm = 2.0f * (float)IN_F * __logf(sg);
        const float pw = pisn[n];
        const v8f dw = tn ? accW1 : accW0;
        const v8f dc = tn ? accC1 : accC0;
#pragma unroll
        for (int e = 0; e < 8; ++e) {
            const int m = row_base + e;
            const float l  = dw[e] + bs;
            const float sq = xnorm[m] - 2.0f * dc[e] + cn;
            const float g  = __expf(-sq * inv_s2c);
            res[(size_t)m * OUT_F + n] = l * g;
            partial = fmaf(sq * inv_s2 + logterm, pw, partial);
        }
    }

    // wave32 reduction of the pi-weighted NLL, one atomic per wave
    for (int off = 16; off > 0; off >>= 1) partial += __shfl_down(partial, off, 32);
    if (lane == 0) atomicAdd(nll_out, partial * inv_total);
}

// ---------------------------------------------------------------------------
extern "C" void kernel_launch(void* const* d_in, const int* in_sizes, int n_in,
                              void* d_out, int out_size, void* d_ws, size_t ws_size,
                              hipStream_t stream) {
    const float* x     = (const float*)d_in[0];
    const float* W     = (const float*)d_in[1];
    const float* C     = (const float*)d_in[2];
    const float* sigs  = (const float*)d_in[3];
    const float* pis   = (const float*)d_in[4];

    float* res = (float*)d_out;                 // (BATCH, OUT_F) flat
    float* nll = res + (size_t)BATCH * OUT_F;   // trailing scalar

    float* ws    = (float*)d_ws;
    float* bias  = ws;                   // OUT_F
    float* cnorm = ws + OUT_F;           // OUT_F
    float* xnorm = ws + 2 * OUT_F;       // BATCH
    float* pisn  = ws + 2 * OUT_F + BATCH; // OUT_F

    // (OUT_F + BATCH) waves, 8 waves/block
    const int pre_blocks = (OUT_F + BATCH) / 8;
    fgn_pre<<<pre_blocks, 256, 0, stream>>>(W, C, x, bias, cnorm, xnorm);
    fgn_softmax<<<1, 256, 0, stream>>>(pis, pisn, nll);

    dim3 grid(OUT_F / 64, BATCH / 64);  // (16, 8) = 128 blocks
    fgn_main<<<grid, 256, 0, stream>>>(x, W, C, sigs, bias, cnorm, xnorm, pisn, res, nll);
}